// BidirectionalRNNEncoder_62251255988734
// MI455X (gfx1250) — compile-verified
//
#include <hip/hip_runtime.h>
#include <math.h>

#define SEQ  16384
#define IDIM 1024
#define HDIM 1024

// ---- GEMM tiling (xw = x @ W + b), f32 WMMA 16x16x4, TDM-fed LDS tiles ----
#define BM 64
#define BN 32
#define KC 32

// ---- recurrence config ----
#define NWG   16                 // workgroups per direction
#define NCOL  (HDIM / NWG)       // 64 columns of U per workgroup (LDS-resident)
#define RPAD  2
#define RSTR  (HDIM + RPAD)      // padded row stride (floats) -> conflict-free float2 reads

typedef __attribute__((ext_vector_type(2))) float v2f;
typedef __attribute__((ext_vector_type(8))) float v8f;
typedef __attribute__((ext_vector_type(4))) unsigned int v4u;
typedef __attribute__((ext_vector_type(8))) int v8i_t;
typedef __attribute__((ext_vector_type(4))) int v4i_t;

// ---------------------------------------------------------------------
// Issue one 2D TDM tile load: global (tile_d1 rows x tile_d0 elems, f32,
// row stride = stride_elems) -> LDS at lds_off, with 1-DWORD padding after
// every 32 DWORDs (so LDS row stride = tile_d0 + 1 floats).
// D# bit layout per CDNA5 ISA 8.3/8.4. Tensor dims == tile dims (never OOB).
// 6-arg builtin form (clang-23 / therock-10.0 headers).
// ---------------------------------------------------------------------
__device__ __forceinline__ void tdm_load_tile_f32(
    unsigned lds_off, const float* gptr,
    unsigned tile_d0, unsigned tile_d1, unsigned long long stride_elems)
{
  unsigned long long ga = (unsigned long long)(uintptr_t)gptr;
  v4u g0;
  g0[0] = 1u;                                              // count=1, load, normal
  g0[1] = lds_off;                                         // lds_addr (bytes)
  g0[2] = (unsigned)(ga & 0xFFFFFFFFu);                    // global_addr[31:0]
  g0[3] = (unsigned)((ga >> 32) & 0x01FFFFFFu) | (2u << 30); // ga[56:32] | type=2

  v8i_t g1;
  g1[0] = (int)((2u << 16) |                               // data_size = 4B
                (1u << 20) |                               // pad_enable
                (4u << 22));                               // pad_interval = 32 DWORDs (amount = 1 DWORD)
  g1[1] = (int)((tile_d0 & 0xFFFFu) << 16);                // tensor_dim0[15:0] (atomic_barrier=0)
  g1[2] = (int)(((tile_d0 >> 16) & 0xFFFFu) | ((tile_d1 & 0xFFFFu) << 16)); // td0 hi | td1 lo
  g1[3] = (int)(((tile_d1 >> 16) & 0xFFFFu) | ((tile_d0 & 0xFFFFu) << 16)); // td1 hi | tile_dim0
  g1[4] = (int)(tile_d1 & 0xFFFFu);                        // tile_dim1 (tile_dim2 = 0 -> 2D)
  g1[5] = (int)(unsigned)(stride_elems & 0xFFFFFFFFu);     // tensor_dim0_stride[31:0]
  g1[6] = (int)(unsigned)((stride_elems >> 32) & 0xFFFFu); // stride[47:32] | dim1_stride lo = 0
  g1[7] = 0;                                               // dim1_stride hi = 0

  v4i_t z4 = {0, 0, 0, 0};                                 // groups 2/3: unused for 2D tile
  v8i_t z8 = {0, 0, 0, 0, 0, 0, 0, 0};                     // trailing group: unused
  __builtin_amdgcn_tensor_load_to_lds(g0, g1, z4, z4, z8, 0);
}

// =====================================================================
// Kernel 1: xw[t][j] = x[row(t)] @ W + b   (both directions via blockIdx.z)
// TDM double-buffered LDS tiles; 8 waves x one 16x16 f32 WMMA accumulator.
// dir==1: A tile loaded in ascending memory order, A-row index reversed.
// =====================================================================
__global__ __launch_bounds__(256) void xw_bias_gemm(
    const float* __restrict__ x,
    const float* __restrict__ Wf, const float* __restrict__ bf,
    const float* __restrict__ Wb, const float* __restrict__ bb,
    float* __restrict__ out)
{
  __shared__ __align__(16) float As[2][BM][KC + 1];  // TDM pad gives the +1
  __shared__ __align__(16) float Bs[2][KC][BN + 1];

  const int dir = blockIdx.z;
  const float* __restrict__ W    = dir ? Wb : Wf;
  const float* __restrict__ bias = dir ? bb : bf;
  float* __restrict__ o = out + (size_t)dir * SEQ * HDIM;

  const int n0  = blockIdx.x * BN;
  const int t0  = blockIdx.y * BM;
  const int tid = threadIdx.x;
  const int lane = tid & 31;
  const int wave = tid >> 5;
  const int wtm  = (wave & 3) * 16;   // wave row offset in [0,BM)
  const int wtn  = (wave >> 2) * 16;  // wave col offset in [0,BN)
  const int half = lane >> 4;         // 0: lanes 0-15, 1: lanes 16-31
  const int lr   = lane & 15;

  // first memory row of the A tile (ascending order even for dir==1)
  const int amem0 = dir ? (SEQ - BM - t0) : t0;
  // A-fragment row inside the LDS tile for matrix row M = wtm+lr
  const int arow  = dir ? (BM - 1 - (wtm + lr)) : (wtm + lr);

  const unsigned ldsA0 = (unsigned)(unsigned long long)(uintptr_t)&As[0][0][0];
  const unsigned ldsA1 = (unsigned)(unsigned long long)(uintptr_t)&As[1][0][0];
  const unsigned ldsB0 = (unsigned)(unsigned long long)(uintptr_t)&Bs[0][0][0];
  const unsigned ldsB1 = (unsigned)(unsigned long long)(uintptr_t)&Bs[1][0][0];

  // accumulator pre-loaded with the bias (same bias for every row of a column)
  v8f c;
  {
    float bv = bias[n0 + wtn + lr];
#pragma unroll
    for (int r = 0; r < 8; ++r) c[r] = bv;
  }

  // prologue: DMA chunk 0 into buffer 0
  if (wave == 0) {
    tdm_load_tile_f32(ldsA0, x + (size_t)amem0 * IDIM, KC, BM, IDIM);
    tdm_load_tile_f32(ldsB0, W + (size_t)0 * HDIM + n0, BN, KC, HDIM);
  }

  int buf = 0;
  for (int k0 = 0; k0 < IDIM; k0 += KC) {
    if (wave == 0) {
      if (k0 + KC < IDIM) {
        // overlap: DMA chunk i+1 into the spare buffer while chunk i computes
        const unsigned la = buf ? ldsA0 : ldsA1;
        const unsigned lb = buf ? ldsB0 : ldsB1;
        tdm_load_tile_f32(la, x + (size_t)amem0 * IDIM + (k0 + KC), KC, BM, IDIM);
        tdm_load_tile_f32(lb, W + (size_t)(k0 + KC) * HDIM + n0, BN, KC, HDIM);
        __builtin_amdgcn_s_wait_tensorcnt(2);  // in-order: chunk i complete
      } else {
        __builtin_amdgcn_s_wait_tensorcnt(0);
      }
    }
    __syncthreads();                           // publish chunk i LDS to all waves

#pragma unroll
    for (int kc = 0; kc < KC; kc += 4) {
      // A 16x4 frag: VGPR0={K+0|K+2}, VGPR1={K+1|K+3} by lane half (ISA 7.12.2)
      v2f a, b;
      a.x = As[buf][arow][kc + 2 * half + 0];
      a.y = As[buf][arow][kc + 2 * half + 1];
      // B 4x16 frag: same K striping, row (=K) across lanes within a VGPR
      b.x = Bs[buf][kc + 2 * half + 0][wtn + lr];
      b.y = Bs[buf][kc + 2 * half + 1][wtn + lr];
      c = __builtin_amdgcn_wmma_f32_16x16x4_f32(
              /*neg_a=*/false, a, /*neg_b=*/false, b,
              /*c_mod=*/(short)0, c, /*reuse_a=*/false, /*reuse_b=*/false);
    }
    __syncthreads();    // all waves done with buf -> safe to DMA into it next iter
    buf ^= 1;
  }

  // C/D layout: VGPR r -> M=r (lanes 0-15) / M=r+8 (lanes 16-31), N=lane%16
#pragma unroll
  for (int r = 0; r < 8; ++r) {
    int m = r + half * 8;
    o[(size_t)(t0 + wtm + m) * HDIM + (n0 + wtn + lr)] = c[r];
  }
}

// =====================================================================
// Kernel 2: reset h double-buffers (h0 = 0) and the arrival counters.
// =====================================================================
__global__ void rnn_init(float* hbuf, unsigned int* cnt)
{
  int i = blockIdx.x * blockDim.x + threadIdx.x;
  if (i < 2 * 2 * HDIM) hbuf[i] = 0.0f;
  if (i < 2) cnt[i] = 0u;
}

// =====================================================================
// Kernel 3: persistent recurrence. 32 workgroups (2 dirs x 16), each holds a
// 1024x64 column slab of U transposed in LDS (256 KB of the 320 KB WGP LDS).
// Per step: stage h (4 KB, L2-resident), k-split partial dots from LDS,
// reduce, tanh, write h chunk + output in place, then a device-scope
// release/acquire counter barrier (double-buffered h => one barrier/step).
// =====================================================================
__global__ __launch_bounds__(256) void rnn_scan(
    const float* __restrict__ Uf, const float* __restrict__ Ub,
    float* __restrict__ out, float* __restrict__ hbuf,
    unsigned int* __restrict__ cnt)
{
  extern __shared__ float smem[];
  float* Ul  = smem;                 // NCOL * RSTR floats (transposed U slab)
  float* hl  = Ul + NCOL * RSTR;     // HDIM floats (current h)
  float* red = hl + HDIM;            // 256 floats (k-split partials)

  const int wg  = blockIdx.x;        // 0..31
  const int dir = wg >> 4;
  const int grp = wg & (NWG - 1);
  const int tid = threadIdx.x;
  const int jbase = grp * NCOL;
  const float* __restrict__ U = dir ? Ub : Uf;
  float* __restrict__ o  = out  + (size_t)dir * SEQ * HDIM;
  float* __restrict__ hb = hbuf + dir * 2 * HDIM;
  unsigned int* c = cnt + dir;

  // one-time transposed load of our U column slab: Ul[jl][k] = U[k][jbase+jl]
  for (int i = tid; i < NCOL * HDIM; i += 256) {
    int k  = i >> 6;                 // i / NCOL (NCOL == 64)
    int jl = i & 63;
    Ul[jl * RSTR + k] = U[(size_t)k * HDIM + (jbase + jl)];
  }

  const int jl = tid & 63;           // output column within slab
  const int kg = tid >> 6;           // 0..3, k-range [kg*256, kg*256+256)
  const float2* up = (const float2*)(Ul + jl * RSTR) + kg * 128;

  for (int t = 0; t < SEQ; ++t) {
    const float* hin  = hb + (t & 1) * HDIM;
    float*       hout = hb + ((t + 1) & 1) * HDIM;

    __syncthreads();                           // hl reuse guard
    for (int i = tid; i < HDIM; i += 256) hl[i] = hin[i];
    __syncthreads();

    float acc = 0.0f;
    const float2* hp = (const float2*)hl + kg * 128;
#pragma unroll 4
    for (int i = 0; i < 128; ++i) {
      float2 u = up[i], h = hp[i];
      acc += u.x * h.x + u.y * h.y;
    }
    red[tid] = acc;
    __syncthreads();

    if (tid < NCOL) {
      float s = red[tid] + red[tid + 64] + red[tid + 128] + red[tid + 192];
      size_t idx = (size_t)t * HDIM + (jbase + tid);
      float hn = tanhf(o[idx] + s);            // o[idx] currently holds xw[t][j]
      o[idx] = hn;                             // in-place: becomes the output
      hout[jbase + tid] = hn;
      __threadfence();                         // make h chunk agent-visible
    }
    __syncthreads();

    if (tid == 0) {
      __hip_atomic_fetch_add(c, 1u, __ATOMIC_RELEASE, __HIP_MEMORY_SCOPE_AGENT);
      const unsigned int target = (unsigned int)NWG * (unsigned int)(t + 1);
      while (__hip_atomic_load(c, __ATOMIC_ACQUIRE, __HIP_MEMORY_SCOPE_AGENT) < target)
        __builtin_amdgcn_s_sleep(1);
    }
    __syncthreads();
  }
}

// =====================================================================
extern "C" void kernel_launch(void* const* d_in, const int* in_sizes, int n_in,
                              void* d_out, int out_size, void* d_ws, size_t ws_size,
                              hipStream_t stream)
{
  const float* x  = (const float*)d_in[0];
  const float* Wf = (const float*)d_in[1];
  const float* Uf = (const float*)d_in[2];
  const float* bf = (const float*)d_in[3];
  const float* Wb = (const float*)d_in[4];
  const float* Ub = (const float*)d_in[5];
  const float* bb = (const float*)d_in[6];
  float* out = (float*)d_out;

  float*        hbuf = (float*)d_ws;                              // 2 dirs x 2 phases x HDIM
  unsigned int* cnt  = (unsigned int*)((char*)d_ws + sizeof(float) * 2 * 2 * HDIM);

  // Phase 1: both input projections (WMMA f32, TDM-fed), xw staged in-place in d_out.
  dim3 ggrid(HDIM / BN, SEQ / BM, 2);
  xw_bias_gemm<<<ggrid, 256, 0, stream>>>(x, Wf, bf, Wb, bb, out);

  // Phase 2a: reset h buffers + barrier counters (deterministic across replays).
  rnn_init<<<16, 256, 0, stream>>>(hbuf, cnt);

  // Phase 2b: persistent scan, 2 dirs x 16 WGs, 256 KB+ dynamic LDS each.
  const size_t smem_bytes = (size_t)(NCOL * RSTR + HDIM + 256) * sizeof(float);
  (void)hipFuncSetAttribute(reinterpret_cast<const void*>(rnn_scan),
                            hipFuncAttributeMaxDynamicSharedMemorySize, (int)smem_bytes);
  rnn_scan<<<2 * NWG, 256, smem_bytes, stream>>>(Uf, Ub, out, hbuf, cnt);
}